// mPD_loss_2_56418690401091
// MI455X (gfx1250) — compile-verified
//
#include <hip/hip_runtime.h>
#include <math.h>

typedef __attribute__((ext_vector_type(2))) float v2f;
typedef __attribute__((ext_vector_type(8))) float v8f;

// Per-element Cartesian difference: deformed(3D + deformation) - original(2D)
__device__ __forceinline__ void calc_d(float r3, float t3, float p3,
                                       float r2, float t2, float p2,
                                       float dt, float dp,
                                       float& dx, float& dy, float& dz) {
    float th3 = t3 + dt, ph3 = p3 + dp;
    float s3  = __sinf(th3), c3  = __cosf(th3);
    float sp3 = __sinf(ph3), cp3 = __cosf(ph3);
    float s2  = __sinf(t2),  c2  = __cosf(t2);
    float sp2 = __sinf(p2),  cp2 = __cosf(p2);
    float a = r3 * s3, b = r2 * s2;
    dx = a * cp3 - b * cp2;
    dy = a * sp3 - b * sp2;
    dz = r3 * c3 - r2 * c2;
}

__global__ void zero_out_kernel(float* out) {
    if (threadIdx.x == 0) out[0] = 0.0f;
}

// Pass 1: per-(batch,span) channel sums of d -> ws[(b*spans+s)*3 + c]
__global__ void span_sums_kernel(const float* __restrict__ sph3,
                                 const float* __restrict__ sph2,
                                 const float* __restrict__ def,
                                 float* __restrict__ ws,
                                 int N, int spans, int totalWaves) {
    int wave = (int)((blockIdx.x * blockDim.x + threadIdx.x) >> 5);
    if (wave >= totalWaves) return;
    int lane = threadIdx.x & 31;
    int b = wave / spans, s = wave % spans;
    int spanLen = N / spans;

    const float* r3p = sph3 + (size_t)b * 3 * N;
    const float* t3p = r3p + N;
    const float* p3p = r3p + 2 * N;
    const float* r2p = sph2 + (size_t)b * 3 * N;
    const float* t2p = r2p + N;
    const float* p2p = r2p + 2 * N;
    const float* dtp = def + (size_t)b * 2 * N;
    const float* dpp = dtp + N;

    int base = s * spanLen;
    float sx = 0.f, sy = 0.f, sz = 0.f;
    int iters = spanLen / 128;  // 32 lanes * float4
    for (int i = 0; i < iters; ++i) {
        int n = base + (i * 32 + lane) * 4;
        float4 r3 = *(const float4*)(r3p + n);
        float4 t3 = *(const float4*)(t3p + n);
        float4 p3 = *(const float4*)(p3p + n);
        float4 r2 = *(const float4*)(r2p + n);
        float4 t2 = *(const float4*)(t2p + n);
        float4 p2 = *(const float4*)(p2p + n);
        float4 dt = *(const float4*)(dtp + n);
        float4 dp = *(const float4*)(dpp + n);
        float dx, dy, dz;
        calc_d(r3.x, t3.x, p3.x, r2.x, t2.x, p2.x, dt.x, dp.x, dx, dy, dz); sx += dx; sy += dy; sz += dz;
        calc_d(r3.y, t3.y, p3.y, r2.y, t2.y, p2.y, dt.y, dp.y, dx, dy, dz); sx += dx; sy += dy; sz += dz;
        calc_d(r3.z, t3.z, p3.z, r2.z, t2.z, p2.z, dt.z, dp.z, dx, dy, dz); sx += dx; sy += dy; sz += dz;
        calc_d(r3.w, t3.w, p3.w, r2.w, t2.w, p2.w, dt.w, dp.w, dx, dy, dz); sx += dx; sy += dy; sz += dz;
    }
    for (int off = 16; off > 0; off >>= 1) {
        sx += __shfl_xor(sx, off, 32);
        sy += __shfl_xor(sy, off, 32);
        sz += __shfl_xor(sz, off, 32);
    }
    if (lane == 0) {
        float* o = ws + (size_t)(b * spans + s) * 3;
        o[0] = sx; o[1] = sy; o[2] = sz;
    }
}

// WMMA-based inclusive scan of a 256-element chunk stored as B-operand tiles.
// X[i][j] = d_{i+16j}. Z = L*X gives within-column prefixes; column sums live in
// row 15 (z[7], lanes 16-31); a 4-step lane scan produces the cross-column offsets.
// Returns updated carry; accumulates sum(|Y|) into acc.
__device__ __forceinline__ float chunk_scan(const v2f a[4], const v2f bm[4],
                                            float carry, float& acc,
                                            int lane, int c) {
    v8f z = {};
    z = __builtin_amdgcn_wmma_f32_16x16x4_f32(false, a[0], false, bm[0], (short)0, z, false, false);
    z = __builtin_amdgcn_wmma_f32_16x16x4_f32(false, a[1], false, bm[1], (short)0, z, false, false);
    z = __builtin_amdgcn_wmma_f32_16x16x4_f32(false, a[2], false, bm[2], (short)0, z, false, false);
    z = __builtin_amdgcn_wmma_f32_16x16x4_f32(false, a[3], false, bm[3], (short)0, z, false, false);

    float scol = __shfl(z[7], 16 + c, 32);      // s_j = column-j total, j = lane%16
    float inc = scol;                            // inclusive scan over the 16 columns
    #pragma unroll
    for (int off = 1; off < 16; off <<= 1) {
        float u = __shfl(inc, (lane - off) & 31, 32);
        if (c >= off) inc += u;
    }
    float pre  = inc - scol;                     // exclusive cross-column prefix
    float addv = carry + pre;
    #pragma unroll
    for (int v = 0; v < 8; ++v) acc += fabsf(z[v] + addv);
    float total = __shfl(inc, 15, 32);           // whole-chunk total
    return carry + total;
}

// Pass 2: per-(batch,span) local scan with carry-in from pass-1 sums; abs-reduce.
__global__ void scan_loss_kernel(const float* __restrict__ o3,
                                 const float* __restrict__ sph3,
                                 const float* __restrict__ o2,
                                 const float* __restrict__ sph2,
                                 const float* __restrict__ def,
                                 const float* __restrict__ ws,
                                 float* __restrict__ out,
                                 int N, int spans, float invNp1, int totalWaves) {
    int wave = (int)((blockIdx.x * blockDim.x + threadIdx.x) >> 5);
    if (wave >= totalWaves) return;
    int lane = threadIdx.x & 31;
    int b = wave / spans, s = wave % spans;
    int spanLen = N / spans;
    int chunks  = spanLen >> 8;
    int c = lane & 15;      // column (N index) of this lane
    int h = lane >> 4;      // lane half

    const float* r3p = sph3 + (size_t)b * 3 * N;
    const float* t3p = r3p + N;
    const float* p3p = r3p + 2 * N;
    const float* r2p = sph2 + (size_t)b * 3 * N;
    const float* t2p = r2p + N;
    const float* p2p = r2p + 2 * N;
    const float* dtp = def + (size_t)b * 2 * N;
    const float* dpp = dtp + N;

    // carry-in: origin difference + sums of all preceding spans
    float carx = o3[b * 3 + 0] - o2[b * 3 + 0];
    float cary = o3[b * 3 + 1] - o2[b * 3 + 1];
    float carz = o3[b * 3 + 2] - o2[b * 3 + 2];
    float acc = 0.f;
    if (s == 0 && lane == 0)  // the k=0 (origin) term of the mean
        acc = fabsf(carx) + fabsf(cary) + fabsf(carz);
    for (int sp = 0; sp < s; ++sp) {
        const float* o = ws + (size_t)(b * spans + sp) * 3;
        carx += o[0]; cary += o[1]; carz += o[2];
    }

    // A operand = lower-triangular ones matrix L, tile t = L[:, 4t:4t+4].
    // A layout (f32 16x4): lane holds row M = lane%16; V0 -> K = 4t+2h, V1 -> K = 4t+2h+1.
    v2f a[4];
    #pragma unroll
    for (int t = 0; t < 4; ++t) {
        int k0 = 4 * t + 2 * h;
        a[t].x = (k0     <= c) ? 1.0f : 0.0f;
        a[t].y = (k0 + 1 <= c) ? 1.0f : 0.0f;
    }

    for (int k = 0; k < chunks; ++k) {
        int chunkStart = s * spanLen + k * 256;
        v2f bx[4], by[4], bz[4];
        #pragma unroll
        for (int t = 0; t < 4; ++t) {
            // B layout (f32 4x16): V0 -> row 4t+2h, V1 -> row 4t+1+2h, col = lane%16.
            // Rows (4t+2h, 4t+2h+1) at col c are elements e0, e0+1 -> one aligned float2.
            int n = chunkStart + 4 * t + 2 * h + 16 * c;
            float2 r3 = *(const float2*)(r3p + n);
            float2 t3 = *(const float2*)(t3p + n);
            float2 p3 = *(const float2*)(p3p + n);
            float2 r2 = *(const float2*)(r2p + n);
            float2 t2 = *(const float2*)(t2p + n);
            float2 p2 = *(const float2*)(p2p + n);
            float2 dt = *(const float2*)(dtp + n);
            float2 dp = *(const float2*)(dpp + n);
            float dx, dy, dz;
            calc_d(r3.x, t3.x, p3.x, r2.x, t2.x, p2.x, dt.x, dp.x, dx, dy, dz);
            bx[t].x = dx; by[t].x = dy; bz[t].x = dz;
            calc_d(r3.y, t3.y, p3.y, r2.y, t2.y, p2.y, dt.y, dp.y, dx, dy, dz);
            bx[t].y = dx; by[t].y = dy; bz[t].y = dz;
        }
        carx = chunk_scan(a, bx, carx, acc, lane, c);
        cary = chunk_scan(a, by, cary, acc, lane, c);
        carz = chunk_scan(a, bz, carz, acc, lane, c);
    }

    for (int off = 16; off > 0; off >>= 1)
        acc += __shfl_xor(acc, off, 32);
    if (lane == 0)
        atomicAdd(out, acc * invNp1);
}

extern "C" void kernel_launch(void* const* d_in, const int* in_sizes, int n_in,
                              void* d_out, int out_size, void* d_ws, size_t ws_size,
                              hipStream_t stream) {
    const float* o3   = (const float*)d_in[0];   // (B,3,1)
    const float* sph3 = (const float*)d_in[1];   // (B,3,N)
    const float* o2   = (const float*)d_in[2];   // (B,3,1)
    const float* sph2 = (const float*)d_in[3];   // (B,3,N)
    const float* def  = (const float*)d_in[4];   // (B,2,N)
    float* out = (float*)d_out;
    float* ws  = (float*)d_ws;

    int B = in_sizes[0] / 3;
    int N = in_sizes[1] / (3 * B);               // 16384
    int spans = N / 1024;                        // span = 1024 elems = 4 WMMA chunks
    if (spans < 1) spans = 1;
    int totalWaves = B * spans;                  // 4096 waves
    float invNp1 = 1.0f / (float)(N + 1);

    int threads = 256;
    int blocks = (totalWaves * 32 + threads - 1) / threads;

    zero_out_kernel<<<1, 32, 0, stream>>>(out);
    span_sums_kernel<<<blocks, threads, 0, stream>>>(sph3, sph2, def, ws, N, spans, totalWaves);
    scan_loss_kernel<<<blocks, threads, 0, stream>>>(o3, sph3, o2, sph2, def, ws, out,
                                                     N, spans, invNp1, totalWaves);
}